// GNN_5205500362786
// MI455X (gfx1250) — compile-verified
//
#include <hip/hip_runtime.h>
#include <hip/hip_bf16.h>

typedef __attribute__((ext_vector_type(2))) float v2f;
typedef __attribute__((ext_vector_type(4))) float v4f;
typedef __attribute__((ext_vector_type(8))) float v8f;

#define D 128            // feature dim (both layers)
#define EPS 1e-12f
#define LDS_STRIDE 260   // 256 + pad: 260%64==4 -> conflict-free ds_load_b64, 16B-aligned rows

// ---------------------------------------------------------------------------
// Stage A/C: sparse SpMM with fused L1 row-normalization.
//   out[row][:] = ( sum_c mask[row][c] * feats[map(c)][:] ) / max(sum_c |mask[row][c]|, eps)
// One wave (32 lanes) per row. Scans the mask row with non-temporal b128 loads
// (mask is read-once: keep the 192MB L2 for the feature gather working set).
// Nonzeros (~11/row) are compacted with ballot+shuffle; each hit gathers one
// 512B feature row (L2-resident).
// ---------------------------------------------------------------------------
__global__ void spmm_l1_kernel(const float* __restrict__ mask, int C,
                               const float* __restrict__ feats,
                               const int* __restrict__ col_map,   // may be null (identity)
                               float* __restrict__ out)
{
    const int lane = threadIdx.x & 31;
    const int wave = threadIdx.x >> 5;
    const int row  = blockIdx.x * (blockDim.x >> 5) + wave;

    const v4f* mrow = (const v4f*)(mask + (size_t)row * C);
    const int nvec = C >> 2;

    v4f acc = (v4f)(0.0f);
    float rowsum = 0.f;

    for (int base = 0; base < nvec; base += 32) {
        v4f v = __builtin_nontemporal_load(&mrow[base + lane]);   // TH=NT stream
        rowsum += fabsf(v.x) + fabsf(v.y) + fabsf(v.z) + fabsf(v.w);
        bool nz = (v.x != 0.f) | (v.y != 0.f) | (v.z != 0.f) | (v.w != 0.f);
        unsigned long long bal = __ballot(nz);          // wave32: low 32 bits
        while (bal) {
            int src = __ffsll(bal) - 1;
            bal &= bal - 1;
            float vv0 = __shfl(v.x, src, 32);
            float vv1 = __shfl(v.y, src, 32);
            float vv2 = __shfl(v.z, src, 32);
            float vv3 = __shfl(v.w, src, 32);
            float vv[4] = {vv0, vv1, vv2, vv3};
            int col0 = (base + src) << 2;
            #pragma unroll
            for (int i = 0; i < 4; ++i) {
                if (vv[i] != 0.f) {                     // uniform across wave
                    int col  = col0 + i;
                    int node = col_map ? col_map[col] : col;
                    v4f fr = ((const v4f*)(feats + (size_t)node * D))[lane];
                    acc += vv[i] * fr;
                }
            }
        }
    }
    // full-wave reduction of the L1 row sum
    #pragma unroll
    for (int off = 16; off >= 1; off >>= 1) rowsum += __shfl_xor(rowsum, off, 32);
    float inv = 1.0f / fmaxf(rowsum, EPS);

    ((v4f*)(out + (size_t)row * D))[lane] = acc * inv;
}

// ---------------------------------------------------------------------------
// Stage B/D: dense layer  out = l2norm_rows( elu( [self | agg] @ W ) )
// One wave per 16-row tile; fp32 WMMA 16x16x4, K=256 (64 steps), 8 N-tiles.
// A-fragments staged through padded LDS; B-fragments read from L2-resident W.
// ISA layout (32-bit): A lanes0-15 K={kb,kb+1}, lanes16-31 K={kb+2,kb+3};
// C/D: VGPR j, lane-half h -> row j+8h, lane%16 -> col.
// ---------------------------------------------------------------------------
__global__ void gnn_layer_wmma(const float* __restrict__ selfFeats,
                               const int*   __restrict__ self_idx,
                               const float* __restrict__ agg,
                               const float* __restrict__ W,     // [2*D][D] row-major
                               float* __restrict__ out)
{
    __shared__ float X[16 * LDS_STRIDE];
    const int lane = threadIdx.x;          // blockDim.x == 32
    const int row0 = blockIdx.x * 16;

    // Stage the 16x256 concat tile: [0..127]=selfFeats[self_idx[r]], [128..255]=agg[r]
    for (int r = 0; r < 16; ++r) {
        int g = row0 + r;
        int sidx = self_idx[g];
        v4f a = ((const v4f*)(selfFeats + (size_t)sidx * D))[lane];
        v4f b = ((const v4f*)(agg + (size_t)g * D))[lane];
        ((v4f*)(X + r * LDS_STRIDE))[lane]     = a;
        ((v4f*)(X + r * LDS_STRIDE + D))[lane] = b;
    }
    __syncthreads();

    v8f acc[8] = {};                       // 8 N-tiles of 16x16 f32
    const int half = lane >> 4;            // 0 | 1
    const int lm   = lane & 15;

    for (int kk = 0; kk < 64; ++kk) {
        const int kb = kk * 4 + half * 2;  // this lane's K pair
        v2f a;
        a.x = X[lm * LDS_STRIDE + kb];
        a.y = X[lm * LDS_STRIDE + kb + 1];
        #pragma unroll
        for (int nt = 0; nt < 8; ++nt) {
            v2f b;
            b.x = W[(size_t)kb * D + nt * 16 + lm];
            b.y = W[(size_t)(kb + 1) * D + nt * 16 + lm];
            acc[nt] = __builtin_amdgcn_wmma_f32_16x16x4_f32(
                false, a, false, b, (short)0, acc[nt], false, false);
        }
    }

    // Epilogue: ELU, then L2-normalize each output row (16 lanes own one row).
    #pragma unroll
    for (int j = 0; j < 8; ++j) {
        float ss = 0.f;
        #pragma unroll
        for (int nt = 0; nt < 8; ++nt) {
            float x = acc[nt][j];
            float y = x > 0.f ? x : expm1f(x);     // elu, alpha=1
            acc[nt][j] = y;
            ss += y * y;
        }
        ss += __shfl_xor(ss, 1, 32);
        ss += __shfl_xor(ss, 2, 32);
        ss += __shfl_xor(ss, 4, 32);
        ss += __shfl_xor(ss, 8, 32);
        float inv = 1.0f / fmaxf(sqrtf(ss), EPS);
        int rowM = row0 + j + 8 * half;
        #pragma unroll
        for (int nt = 0; nt < 8; ++nt)
            out[(size_t)rowM * D + nt * 16 + lm] = acc[nt][j] * inv;
    }
}

// ---------------------------------------------------------------------------
// Inputs (setup_inputs order): raw_features, mask1, mask2, W1, W2,
//                              idx_embed0, self_idx1, self_idx2
// ---------------------------------------------------------------------------
extern "C" void kernel_launch(void* const* d_in, const int* in_sizes, int n_in,
                              void* d_out, int out_size, void* d_ws, size_t ws_size,
                              hipStream_t stream) {
    const float* raw        = (const float*)d_in[0];
    const float* mask1      = (const float*)d_in[1];
    const float* mask2      = (const float*)d_in[2];
    const float* W1         = (const float*)d_in[3];
    const float* W2         = (const float*)d_in[4];
    const int*   idx_embed0 = (const int*)d_in[5];
    const int*   self_idx1  = (const int*)d_in[6];
    const int*   self_idx2  = (const int*)d_in[7];
    float* out = (float*)d_out;

    const int ROWS1 = 8192, COLS1 = 24576, B = 1024;

    float* agg1 = (float*)d_ws;                 // [8192 x 128]
    float* h1   = agg1 + (size_t)ROWS1 * D;     // [8192 x 128]
    float* agg2 = h1   + (size_t)ROWS1 * D;     // [1024 x 128]

    // Layer 1: agg1 = l1(mask1) @ raw[idx_embed0]   (gather fused via col_map)
    spmm_l1_kernel<<<ROWS1 / 8, 256, 0, stream>>>(mask1, COLS1, raw, idx_embed0, agg1);
    // h1 = l2(elu([raw[self_idx1] | agg1] @ W1))
    gnn_layer_wmma<<<ROWS1 / 16, 32, 0, stream>>>(raw, self_idx1, agg1, W1, h1);
    // Layer 2: agg2 = l1(mask2) @ h1
    spmm_l1_kernel<<<B / 8, 256, 0, stream>>>(mask2, ROWS1, h1, nullptr, agg2);
    // h2 = l2(elu([h1[self_idx2] | agg2] @ W2))
    gnn_layer_wmma<<<B / 16, 32, 0, stream>>>(h1, self_idx2, agg2, W2, out);
}